// MessagePassing_57432302682772
// MI455X (gfx1250) — compile-verified
//
#include <hip/hip_runtime.h>

#define D_FEAT 128

// Non-returning, device-scope, relaxed f32 atomic add.
// Result is discarded -> GLOBAL_ATOMIC_ADD_F32 with TH = non-returning
// (fire-and-forget at the L2 atomic units).
__device__ __forceinline__ void atomic_add_f32(float* p, float v) {
    __hip_atomic_fetch_add(p, v, __ATOMIC_RELAXED, __HIP_MEMORY_SCOPE_AGENT);
}

// One edge per wave32: 128 features = 32 lanes x float4.
//  - gather:  one global_load_b128 per lane from x[src]  (L2-resident, 51.2 MB)
//  - scatter: 4 non-returning global_atomic_add_f32 per lane into out[dst]
// Defined FIRST so the disassembly snippet shows this kernel's codegen.
__global__ __launch_bounds__(256) void MessagePassing_scatter_kernel(
    const float* __restrict__ x,
    const int*   __restrict__ src,
    const int*   __restrict__ dst,
    float*       __restrict__ out,
    int n_edges)
{
    const int lane          = threadIdx.x & 31;        // wave32 lane
    const int waveInBlock   = threadIdx.x >> 5;        // 8 waves per 256-thread block
    const int wavesPerBlock = blockDim.x >> 5;

    const long long e = (long long)blockIdx.x * wavesPerBlock + waveInBlock;
    if (e >= n_edges) return;

    const int s = src[e];   // wave-uniform (all lanes hit the same cacheline)
    const int d = dst[e];

    const float4 v = *reinterpret_cast<const float4*>(
        x + (size_t)s * D_FEAT + (size_t)lane * 4);

    float* o = out + (size_t)d * D_FEAT + (size_t)lane * 4;
    atomic_add_f32(o + 0, v.x);
    atomic_add_f32(o + 1, v.y);
    atomic_add_f32(o + 2, v.z);
    atomic_add_f32(o + 3, v.w);
}

// Zero the output accumulator with 128-bit stores.
__global__ __launch_bounds__(256) void zero_f32x4_kernel(float4* __restrict__ out,
                                                         int n4) {
    int i = blockIdx.x * blockDim.x + threadIdx.x;
    if (i < n4) {
        out[i] = make_float4(0.f, 0.f, 0.f, 0.f);
    }
}

extern "C" void kernel_launch(void* const* d_in, const int* in_sizes, int n_in,
                              void* d_out, int out_size, void* d_ws, size_t ws_size,
                              hipStream_t stream) {
    const float* x  = (const float*)d_in[0];
    const int*   ei = (const int*)d_in[1];   // flattened (2, E): row0 = src, row1 = dst

    const int E    = in_sizes[1] / 2;
    const int* src = ei;
    const int* dst = ei + E;

    float* out = (float*)d_out;

    // 1) zero the accumulator (harness poisons d_out with 0xAA)
    const int n4 = out_size / 4;
    {
        const int threads = 256;
        const int blocks  = (n4 + threads - 1) / threads;
        zero_f32x4_kernel<<<blocks, threads, 0, stream>>>((float4*)out, n4);
    }

    // 2) gather + scatter-add, one edge per wave32 (8 edges per 256-thread block)
    {
        const int threads       = 256;
        const int wavesPerBlock = threads / 32;
        const int blocks        = (E + wavesPerBlock - 1) / wavesPerBlock;
        MessagePassing_scatter_kernel<<<blocks, threads, 0, stream>>>(
            x, src, dst, out, E);
    }
}